// GNNNodeAndSceneClassifier_58884001628501
// MI455X (gfx1250) — compile-verified
//
#include <hip/hip_runtime.h>
#include <math.h>

typedef __attribute__((ext_vector_type(2))) float v2f;
typedef __attribute__((ext_vector_type(4))) float v4f;
typedef __attribute__((ext_vector_type(8))) float v8f;

// ---------------------------------------------------------------- utilities

__global__ void zero_kernel(float* __restrict__ p, long long n) {
    long long i = blockIdx.x * (long long)blockDim.x + threadIdx.x;
    long long stride = (long long)gridDim.x * blockDim.x;
    for (; i < n; i += stride) p[i] = 0.0f;
}

__global__ void degree_kernel(const int* __restrict__ dst, float* __restrict__ deg, int E) {
    int i = blockIdx.x * blockDim.x + threadIdx.x;
    int stride = gridDim.x * blockDim.x;
    for (; i < E; i += stride) atomicAdd(&deg[dst[i]], 1.0f);
}

// Scatter-add D-wide rows into accum[dst[e]].
//   GATHER=false: vals indexed by edge id (edge features), NT streaming load
//   GATHER=true : vals indexed by srcIdx[e]
// Each thread moves 4 consecutive floats (16B load + 4 f32 atomics, L2-side).
template <int D, bool GATHER, bool NT>
__global__ void scatter_add_kernel(const float* __restrict__ vals,
                                   const int* __restrict__ srcIdx,
                                   const int* __restrict__ dstIdx,
                                   float* __restrict__ accum,
                                   int E) {
    constexpr int CHUNKS = D >> 2;
    constexpr int CSHIFT = (D == 128) ? 5 : (D == 64) ? 4 : (D == 32) ? 3 : 2;
    long long i = blockIdx.x * (long long)blockDim.x + threadIdx.x;
    const long long total  = (long long)E * CHUNKS;
    const long long stride = (long long)gridDim.x * blockDim.x;
    for (; i < total; i += stride) {
        const int e = (int)(i >> CSHIFT);
        const int c = (int)(i & (CHUNKS - 1));
        const int row = GATHER ? srcIdx[e] : e;
        const v4f* src4 = (const v4f*)(vals + (long long)row * D) + c;
        v4f v;
        if constexpr (NT) v = __builtin_nontemporal_load(src4);
        else              v = *src4;
        float* d = accum + (long long)dstIdx[e] * D + (c << 2);
        atomicAdd(d + 0, v.x);
        atomicAdd(d + 1, v.y);
        atomicAdd(d + 2, v.z);
        atomicAdd(d + 3, v.w);
    }
}

// ------------------------------------------------- WMMA mean->Linear(+ReLU)
// out[m, j] = act( (accum[m,:] / max(deg[m],1)) @ W[K,NOUT] + b[j] )
// 4 waves / block; each wave computes one 16x16 output tile with
// V_WMMA_F32_16X16X4_F32, looping K in steps of 4.
//
// Frag layouts (CDNA5 ISA 7.12.2, 32-bit operands):
//   A 16x4:  lanes 0-15 -> M=lane,  v0=K0 v1=K1 ; lanes 16-31 -> v0=K2 v1=K3
//   B 4x16:  lanes 0-15 -> N=lane,  v0=K0 v1=K1 ; lanes 16-31 -> v0=K2 v1=K3
//   C 16x16: VGPR i: lanes 0-15 -> M=i, lanes 16-31 -> M=i+8 ; N=lane%16
template <int K, int NOUT, bool RELU>
__global__ void wmma_mean_linear_kernel(const float* __restrict__ accum,
                                        const float* __restrict__ deg,
                                        const float* __restrict__ W,
                                        const float* __restrict__ bias,
                                        float* __restrict__ out,
                                        int Nnodes) {
    constexpr int KITER = K / 4;
    const int wave  = threadIdx.x >> 5;
    const int tileM = blockIdx.x * 4 + wave;
    const int tileN = blockIdx.y;
    if (tileM * 16 >= Nnodes) return;           // wave-uniform exit

    const int lane = threadIdx.x & 31;
    const int half = lane >> 4;
    const int lp   = lane & 15;

    const int  m      = tileM * 16 + lp;        // A row this lane loads
    const bool mvalid = (m < Nnodes);
    const int  mc     = mvalid ? m : 0;
    const float inv   = mvalid ? (1.0f / fmaxf(deg[mc], 1.0f)) : 0.0f;

    const int  ncol   = tileN * 16 + lp;        // B column this lane loads
    const bool nvalid = (ncol < NOUT);

    const float* arow = accum + (long long)mc * K;

    // For ragged NOUT (node classifier, NOUT=2, K=16): stage B fragments in
    // registers under ONE exec-guarded region so the WMMA loop is branch-free.
    v2f bfrag[KITER];
    if constexpr (NOUT % 16 != 0) {
#pragma unroll
        for (int t = 0; t < KITER; ++t) { bfrag[t].x = 0.0f; bfrag[t].y = 0.0f; }
        if (nvalid) {
#pragma unroll
            for (int t = 0; t < KITER; ++t) {
                const int ka = t * 4 + 2 * half;
                bfrag[t].x = W[ka       * NOUT + ncol];
                bfrag[t].y = W[(ka + 1) * NOUT + ncol];
            }
        }
    }

    v8f c = {};
#pragma unroll
    for (int t = 0; t < KITER; ++t) {
        const int ka = t * 4 + 2 * half;        // K index for v0; v1 = ka+1
        v2f a, b;
        a.x = arow[ka]     * inv;
        a.y = arow[ka + 1] * inv;
        if constexpr (NOUT % 16 == 0) {
            b.x = W[ka       * NOUT + ncol];    // always in-bounds: unguarded
            b.y = W[(ka + 1) * NOUT + ncol];
        } else {
            b = bfrag[t];
        }
        c = __builtin_amdgcn_wmma_f32_16x16x4_f32(
                /*neg_a=*/false, a, /*neg_b=*/false, b,
                /*c_mod=*/(short)0, c, /*reuse_a=*/false, /*reuse_b=*/false);
    }

    const float bv = nvalid ? bias[ncol] : 0.0f;
#pragma unroll
    for (int i = 0; i < 8; ++i) {
        const int row = tileM * 16 + 8 * half + i;   // C/D layout
        if (row < Nnodes && nvalid) {
            float v = c[i] + bv;
            if (RELU) v = fmaxf(v, 0.0f);
            out[(long long)row * NOUT + ncol] = v;
        }
    }
}

// ------------------------------------------------------------- graph head

// Column sums of h2[N,16] -> gsum[16]
__global__ void colsum16_kernel(const float* __restrict__ h, float* __restrict__ gsum, int N) {
    const int f = threadIdx.x & 15;
    const int grp  = (blockIdx.x * blockDim.x + threadIdx.x) >> 4;
    const int ngrp = (gridDim.x * blockDim.x) >> 4;
    float s = 0.0f;
    for (int n = grp; n < N; n += ngrp) s += h[(long long)n * 16 + f];
    atomicAdd(&gsum[f], s);
}

// graph_label = sigmoid( relu(g @ Wg1 + bg1) @ Wg2 + bg2 ), g = gsum / N
__global__ void graph_head_kernel(const float* __restrict__ gsum,
                                  const float* __restrict__ Wg1, const float* __restrict__ bg1,
                                  const float* __restrict__ Wg2, const float* __restrict__ bg2,
                                  float* __restrict__ out, int N) {
    if (threadIdx.x != 0 || blockIdx.x != 0) return;
    float g[16];
    const float invN = 1.0f / (float)N;
#pragma unroll
    for (int j = 0; j < 16; ++j) g[j] = gsum[j] * invN;
    float z = bg2[0];
#pragma unroll
    for (int i = 0; i < 8; ++i) {
        float t = bg1[i];
#pragma unroll
        for (int j = 0; j < 16; ++j) t += g[j] * Wg1[j * 8 + i];
        t = fmaxf(t, 0.0f);
        z += t * Wg2[i];
    }
    out[0] = 1.0f / (1.0f + expf(-z));
}

// ------------------------------------------------------------------ launch

static inline int clampBlocks(long long work, int threads, int maxBlocks) {
    long long b = (work + threads - 1) / threads;
    if (b > maxBlocks) b = maxBlocks;
    if (b < 1) b = 1;
    return (int)b;
}

extern "C" void kernel_launch(void* const* d_in, const int* in_sizes, int n_in,
                              void* d_out, int out_size, void* d_ws, size_t ws_size,
                              hipStream_t stream) {
    const float* edge_feats = (const float*)d_in[0];
    const float* W0  = (const float*)d_in[1];
    const float* b0  = (const float*)d_in[2];
    const float* W1  = (const float*)d_in[3];
    const float* b1  = (const float*)d_in[4];
    const float* W2  = (const float*)d_in[5];
    const float* b2  = (const float*)d_in[6];
    const float* Wn  = (const float*)d_in[7];
    const float* bn  = (const float*)d_in[8];
    const float* Wg1 = (const float*)d_in[9];
    const float* bg1 = (const float*)d_in[10];
    const float* Wg2 = (const float*)d_in[11];
    const float* bg2 = (const float*)d_in[12];
    const int*   src = (const int*)d_in[13];
    const int*   dst = (const int*)d_in[14];

    const int E = in_sizes[13];                 // 800000
    const int N = (out_size - 1) / 2;           // 50000 (node_label is [N,2], +1 graph label)

    float* out  = (float*)d_out;
    float* gout = out;                          // [1]  graph label
    float* nout = out + 1;                      // [N,2] node labels

    // ---- workspace layout (floats), ~48 MB ----
    float* ws    = (float*)d_ws;
    float* accum = ws;                                  // N*128 (reused at 64/32/16)
    float* h0    = accum + (long long)N * 128;          // N*64
    float* h1    = h0    + (long long)N * 64;           // N*32
    float* h2    = h1    + (long long)N * 32;           // N*16
    float* deg   = h2    + (long long)N * 16;           // N
    float* gsum  = deg   + N;                           // 16
    (void)ws_size; (void)n_in;

    const int T = 256;
    const int tilesM  = (N + 15) / 16;
    const int mBlocks = (tilesM + 3) / 4;               // 4 waves per block

    // degree (shared by every segment-mean)
    zero_kernel<<<clampBlocks(N, T, 4096), T, 0, stream>>>(deg, N);
    degree_kernel<<<clampBlocks(E, T, 65536), T, 0, stream>>>(dst, deg, E);

    // ---- layer 0: mean(edge_feats by dst) @ W0 + b0, relu ----
    zero_kernel<<<clampBlocks((long long)N * 128, T, 65536), T, 0, stream>>>(accum, (long long)N * 128);
    scatter_add_kernel<128, false, true><<<clampBlocks((long long)E * 32, T, 131072), T, 0, stream>>>(
        edge_feats, nullptr, dst, accum, E);
    wmma_mean_linear_kernel<128, 64, true><<<dim3(mBlocks, 64 / 16), 128, 0, stream>>>(
        accum, deg, W0, b0, h0, N);

    // ---- layer 1: mean(h0[src] by dst) @ W1 + b1, relu ----
    zero_kernel<<<clampBlocks((long long)N * 64, T, 65536), T, 0, stream>>>(accum, (long long)N * 64);
    scatter_add_kernel<64, true, false><<<clampBlocks((long long)E * 16, T, 131072), T, 0, stream>>>(
        h0, src, dst, accum, E);
    wmma_mean_linear_kernel<64, 32, true><<<dim3(mBlocks, 32 / 16), 128, 0, stream>>>(
        accum, deg, W1, b1, h1, N);

    // ---- layer 2: mean(h1[src] by dst) @ W2 + b2, relu ----
    zero_kernel<<<clampBlocks((long long)N * 32, T, 65536), T, 0, stream>>>(accum, (long long)N * 32);
    scatter_add_kernel<32, true, false><<<clampBlocks((long long)E * 8, T, 131072), T, 0, stream>>>(
        h1, src, dst, accum, E);
    wmma_mean_linear_kernel<32, 16, true><<<dim3(mBlocks, 1), 128, 0, stream>>>(
        accum, deg, W2, b2, h2, N);

    // ---- graph head ----
    zero_kernel<<<1, 32, 0, stream>>>(gsum, 16);
    colsum16_kernel<<<256, 256, 0, stream>>>(h2, gsum, N);
    graph_head_kernel<<<1, 32, 0, stream>>>(gsum, Wg1, bg1, Wg2, bg2, gout, N);

    // ---- node classifier: mean(h2[src] by dst) @ Wn + bn (no act) ----
    zero_kernel<<<clampBlocks((long long)N * 16, T, 65536), T, 0, stream>>>(accum, (long long)N * 16);
    scatter_add_kernel<16, true, false><<<clampBlocks((long long)E * 4, T, 131072), T, 0, stream>>>(
        h2, src, dst, accum, E);
    wmma_mean_linear_kernel<16, 2, false><<<dim3(mBlocks, 1), 128, 0, stream>>>(
        accum, deg, Wn, bn, nout, N);
}